// KDPointToPointLoss_47038481826616
// MI455X (gfx1250) — compile-verified
//
#include <hip/hip_runtime.h>

typedef float v2f __attribute__((ext_vector_type(2)));
typedef float v8f __attribute__((ext_vector_type(8)));

#define B_SZ      2
#define N_PTS     8192
#define M_PTS     8192
#define SRC_TILES (N_PTS / 16)            // 512
#define TGT_TILES (M_PTS / 16)            // 512
#define MULT      2                        // source tiles per wave
#define NWAVES    (B_SZ * SRC_TILES / MULT)  // 512

// One wave32 per (batch, pair of 16-source-point tiles).
// Score[m,n] = t_m . s_n - 0.5*||t_m||^2  via V_WMMA_F32_16X16X4_F32,
// argmax over m == argmin over m of ||s_n - t_m||^2.
// The -0.5*||t||^2 bias rides in the K=3 slot of the A fragment (B's K=3 is 1).
__global__ __launch_bounds__(32)
void nn_wmma_kernel(const float* __restrict__ src,
                    const float* __restrict__ tgt,
                    float* __restrict__ partial)
{
    const int wave  = blockIdx.x;              // 0..NWAVES-1
    const int b     = wave / (SRC_TILES / MULT);
    const int sPair = wave % (SRC_TILES / MULT);
    const int lane  = threadIdx.x;             // 0..31 (wave32)
    const int half  = lane >> 4;               // 0: K=0,1 ; 1: K=2,3
    const int l16   = lane & 15;

    const float* sBase = src + (size_t)b * N_PTS * 3;
    const float* tBase = tgt + (size_t)b * M_PTS * 3;

    // ---- B fragments: source columns n0, n1 = n0+16; [sx, sy, sz, 1] along K ----
    const int n0 = sPair * (16 * MULT) + l16;
    const int n1 = n0 + 16;
    const float sx0 = sBase[n0 * 3 + 0], sy0 = sBase[n0 * 3 + 1], sz0 = sBase[n0 * 3 + 2];
    const float sx1 = sBase[n1 * 3 + 0], sy1 = sBase[n1 * 3 + 1], sz1 = sBase[n1 * 3 + 2];
    v2f bfrag0, bfrag1;
    bfrag0.x = half ? sz0 : sx0;   bfrag0.y = half ? 1.0f : sy0;
    bfrag1.x = half ? sz1 : sx1;   bfrag1.y = half ? 1.0f : sy1;

    float best0 = -3.402823466e38f, best1 = -3.402823466e38f;
    int   idx0  = 0,                idx1  = 0;

    for (int t = 0; t < TGT_TILES; ++t) {
        // ---- A fragment (shared by both WMMAs): [tx, ty, tz, -0.5*||t||^2] ----
        const int m = t * 16 + l16;
        const float tx = tBase[m * 3 + 0];
        const float ty = tBase[m * 3 + 1];
        const float tz = tBase[m * 3 + 2];
        const float t2 = tx * tx + ty * ty + tz * tz;
        v2f afrag;
        afrag.x = half ? tz : tx;               // K=2 : K=0
        afrag.y = half ? (-0.5f * t2) : ty;     // K=3 : K=1

        v8f c0 = {}, c1 = {};
        // 8 args: (neg_a, A, neg_b, B, c_mod, C, reuse_a, reuse_b)
        v8f d0 = __builtin_amdgcn_wmma_f32_16x16x4_f32(
            false, afrag, false, bfrag0, (short)0, c0, false, false);
        v8f d1 = __builtin_amdgcn_wmma_f32_16x16x4_f32(
            false, afrag, false, bfrag1, (short)0, c1, false, false);

        // Each lane holds target rows (t*16 + half*8 + v) for its source col.
        const int rowBase = t * 16 + half * 8;
        #pragma unroll
        for (int v = 0; v < 8; ++v) {
            const int  idx = rowBase + v;
            const float s0 = d0[v];
            const bool g0  = s0 > best0;        // strict > keeps earliest index
            idx0  = g0 ? idx : idx0;
            best0 = g0 ? s0  : best0;
            const float s1 = d1[v];
            const bool g1  = s1 > best1;
            idx1  = g1 ? idx : idx1;
            best1 = g1 ? s1  : best1;
        }
    }

    // Merge lane-pair halves (rows 0-7 in lane, rows 8-15 in lane^16).
    {
        const float o = __shfl_xor(best0, 16, 32);
        const int   oi = __shfl_xor(idx0, 16, 32);
        if (o > best0 || (o == best0 && oi < idx0)) { best0 = o; idx0 = oi; }
    }
    {
        const float o = __shfl_xor(best1, 16, 32);
        const int   oi = __shfl_xor(idx1, 16, 32);
        if (o > best1 || (o == best1 && oi < idx1)) { best1 = o; idx1 = oi; }
    }

    // Squared errors for this lane's two source points (count once: lower half).
    float err = 0.0f;
    {
        const float bx = tBase[idx0 * 3 + 0], by = tBase[idx0 * 3 + 1], bz = tBase[idx0 * 3 + 2];
        const float dx = sx0 - bx, dy = sy0 - by, dz = sz0 - bz;
        err += dx * dx + dy * dy + dz * dz;
    }
    {
        const float bx = tBase[idx1 * 3 + 0], by = tBase[idx1 * 3 + 1], bz = tBase[idx1 * 3 + 2];
        const float dx = sx1 - bx, dy = sy1 - by, dz = sz1 - bz;
        err += dx * dx + dy * dy + dz * dz;
    }
    if (half) err = 0.0f;

    // In-wave sum: xor masks 1..8 stay within each 16-lane half (upper = 0).
    err += __shfl_xor(err, 1, 32);
    err += __shfl_xor(err, 2, 32);
    err += __shfl_xor(err, 4, 32);
    err += __shfl_xor(err, 8, 32);
    err += __shfl_xor(err, 16, 32);

    if (lane == 0) partial[wave] = err;
}

__global__ __launch_bounds__(256)
void reduce_kernel(const float* __restrict__ partial, float* __restrict__ out)
{
    __shared__ float smem[256];
    const int tid = threadIdx.x;
    float s = 0.0f;
    for (int i = tid; i < NWAVES; i += 256) s += partial[i];  // fixed order
    smem[tid] = s;
    __syncthreads();
    for (int off = 128; off > 0; off >>= 1) {
        if (tid < off) smem[tid] += smem[tid + off];
        __syncthreads();
    }
    if (tid == 0)
        out[0] = smem[0] * (1.0f / (float)(B_SZ * N_PTS * 3));
}

extern "C" void kernel_launch(void* const* d_in, const int* in_sizes, int n_in,
                              void* d_out, int out_size, void* d_ws, size_t ws_size,
                              hipStream_t stream)
{
    (void)in_sizes; (void)n_in; (void)out_size; (void)ws_size;
    const float* src = (const float*)d_in[0];   // [B, N, 3] f32
    const float* tgt = (const float*)d_in[1];   // [B, M, 3] f32
    float* partial = (float*)d_ws;              // NWAVES floats (2 KB)
    float* out     = (float*)d_out;             // scalar f32

    nn_wmma_kernel<<<NWAVES, 32, 0, stream>>>(src, tgt, partial);
    reduce_kernel<<<1, 256, 0, stream>>>(partial, out);
}